// TwoLayerGCN_34591666602851
// MI455X (gfx1250) — compile-verified
//
#include <hip/hip_runtime.h>
#include <math.h>

#define N_NODES 100000
#define N_EDGES 1600000
#define IN_DIM  128
#define HID_DIM 64

typedef float v2f __attribute__((ext_vector_type(2)));
typedef float v8f __attribute__((ext_vector_type(8)));

// ---------------- degree / normalization ----------------

__global__ void k_init_deg(int* degi) {
    int i = blockIdx.x * blockDim.x + threadIdx.x;
    if (i < N_NODES) degi[i] = 1;               // self-loop contributes 1
}

__global__ void k_count_deg(const int* __restrict__ dst, int* degi) {
    int e = blockIdx.x * blockDim.x + threadIdx.x;
    if (e < N_EDGES) atomicAdd(&degi[dst[e]], 1);
}

__global__ void k_dinv(const int* __restrict__ degi, float* __restrict__ dinv) {
    int i = blockIdx.x * blockDim.x + threadIdx.x;
    if (i < N_NODES) dinv[i] = rsqrtf((float)degi[i]);   // deg >= 1 always
}

__global__ void k_zero(float* p, int n) {
    int i = blockIdx.x * blockDim.x + threadIdx.x;
    if (i < n) p[i] = 0.0f;
}

// ---------------- layer 1 GEMM: h = x @ W1  (WMMA f32 16x16x4) ----------------
// Block: 64 threads = 2 waves. Wave w handles row tile (blockIdx*2 + w)*16,
// all 64 output columns (4 accumulators). 3125 blocks * 2 waves * 16 rows = 100000.
//
// W1 is staged in LDS pre-swizzled as K-pairs: lw[kp*STRIDE + n] = {W1[2kp][n], W1[2kp+1][n]}
// so each B fragment is ONE contiguous ds_load_b64 straight into an aligned VGPR pair
// (no 2addr loads, no v_mov repacking). STRIDE has a 16-pair (128B) pad so the two
// lane-halves (K-pairs 1 apart) hit disjoint LDS bank halves: conflict-free b64 loads.

#define LDS_PAIR_STRIDE 80   // 64 pairs + 16 pad

__global__ __launch_bounds__(64) void k_gemm1(const float* __restrict__ x,
                                              const float* __restrict__ W1,
                                              float* __restrict__ h) {
    __shared__ float2 lw[64 * LDS_PAIR_STRIDE];   // 40 KB of the 320 KB LDS
    // cooperative swizzled preload: coalesced global reads, pair-packed LDS writes
    for (int idx = threadIdx.x; idx < 64 * 64; idx += 64) {
        const int kp = idx >> 6;     // K-pair index 0..63
        const int n  = idx & 63;     // output column
        float2 v;
        v.x = W1[(2 * kp + 0) * HID_DIM + n];
        v.y = W1[(2 * kp + 1) * HID_DIM + n];
        lw[kp * LDS_PAIR_STRIDE + n] = v;
    }
    __syncthreads();

    const int wave = threadIdx.x >> 5;
    const int lane = threadIdx.x & 31;
    const int rowTile = (blockIdx.x * 2 + wave) * 16;

    // A fragment addressing (16x4 f32): lanes 0-15 hold K={0,1}, lanes 16-31 K={2,3}
    const int m  = lane & 15;
    const int kh = lane >> 4;        // K-half select
    const int n  = m;                // B / C-D column index

    v8f acc0 = {}; v8f acc1 = {}; v8f acc2 = {}; v8f acc3 = {};

    const float* xrow = x + (size_t)(rowTile + m) * IN_DIM;

    #pragma unroll 4
    for (int k0 = 0; k0 < IN_DIM; k0 += 4) {
        float2 av = *(const float2*)(xrow + k0 + 2 * kh);  // contiguous K pair
        v2f a; a.x = av.x; a.y = av.y;

        const float2* bp = lw + ((k0 >> 1) + kh) * LDS_PAIR_STRIDE + n;
        float2 t; v2f b;

        t = bp[0];  b.x = t.x; b.y = t.y;
        acc0 = __builtin_amdgcn_wmma_f32_16x16x4_f32(false, a, false, b, (short)0, acc0, false, false);
        t = bp[16]; b.x = t.x; b.y = t.y;
        acc1 = __builtin_amdgcn_wmma_f32_16x16x4_f32(false, a, false, b, (short)0, acc1, false, false);
        t = bp[32]; b.x = t.x; b.y = t.y;
        acc2 = __builtin_amdgcn_wmma_f32_16x16x4_f32(false, a, false, b, (short)0, acc2, false, false);
        t = bp[48]; b.x = t.x; b.y = t.y;
        acc3 = __builtin_amdgcn_wmma_f32_16x16x4_f32(false, a, false, b, (short)0, acc3, false, false);
    }

    // C/D layout: VGPR r -> row (r + 8*(lane>=16)), col = lane&15
    const int mb = kh * 8;
    float* hp = h + (size_t)rowTile * HID_DIM;
    #pragma unroll
    for (int r = 0; r < 8; ++r) {
        const int row = mb + r;
        hp[(size_t)row * HID_DIM +  0 + n] = acc0[r];
        hp[(size_t)row * HID_DIM + 16 + n] = acc1[r];
        hp[(size_t)row * HID_DIM + 32 + n] = acc2[r];
        hp[(size_t)row * HID_DIM + 48 + n] = acc3[r];
    }
}

// ---------------- layer 1 scatter: agg1[dst] += norm * h[src] ----------------
// 16 threads per edge, each handles a float4 chunk of the 64 features.

__global__ __launch_bounds__(256) void k_scatter1(const int* __restrict__ src,
                                                  const int* __restrict__ dst,
                                                  const float* __restrict__ dinv,
                                                  const float* __restrict__ h,
                                                  float* agg1) {
    const unsigned long long tid = (unsigned long long)blockIdx.x * 256ull + threadIdx.x;
    const int e = (int)(tid >> 4);
    const int c = (int)(tid & 15);
    if (e < N_EDGES) {
        const int s = src[e], d = dst[e];
        const float w = dinv[s] * dinv[d];
        const float4 v = *(const float4*)(h + (size_t)s * HID_DIM + c * 4);
        float* o = agg1 + (size_t)d * HID_DIM + c * 4;
        atomicAdd(o + 0, w * v.x);
        atomicAdd(o + 1, w * v.y);
        atomicAdd(o + 2, w * v.z);
        atomicAdd(o + 3, w * v.w);
    }
}

// self-loop term + bias + ReLU, in place (agg1 becomes h1)
__global__ void k_relu_bias(float* agg1,
                            const float* __restrict__ h,
                            const float* __restrict__ dinv,
                            const float* __restrict__ b1) {
    int i = blockIdx.x * blockDim.x + threadIdx.x;
    if (i < N_NODES * HID_DIM) {
        const int node = i >> 6;
        const int f = i & 63;
        const float di = dinv[node];
        float v = agg1[i] + di * di * h[i] + b1[f];
        agg1[i] = v > 0.0f ? v : 0.0f;
    }
}

// ---------------- layer 2 matvec: z = h1 @ W2 (64 -> 1), one wave per node ----------------

__global__ __launch_bounds__(256) void k_matvec2(const float* __restrict__ h1,
                                                 const float* __restrict__ W2,
                                                 float* __restrict__ z) {
    const int wid  = (int)((blockIdx.x * 256u + threadIdx.x) >> 5);
    const int lane = threadIdx.x & 31;
    if (wid < N_NODES) {
        const float* row = h1 + (size_t)wid * HID_DIM;
        float s = row[lane] * W2[lane] + row[lane + 32] * W2[lane + 32];
        #pragma unroll
        for (int off = 16; off > 0; off >>= 1) s += __shfl_xor(s, off, 32);
        if (lane == 0) z[wid] = s;
    }
}

__global__ void k_scatter2(const int* __restrict__ src,
                           const int* __restrict__ dst,
                           const float* __restrict__ dinv,
                           const float* __restrict__ z,
                           float* agg2) {
    int e = blockIdx.x * blockDim.x + threadIdx.x;
    if (e < N_EDGES) {
        const int s = src[e], d = dst[e];
        atomicAdd(&agg2[d], dinv[s] * dinv[d] * z[s]);
    }
}

__global__ void k_final(const float* __restrict__ agg2,
                        const float* __restrict__ z,
                        const float* __restrict__ dinv,
                        const float* __restrict__ b2,
                        float* __restrict__ out) {
    int i = blockIdx.x * blockDim.x + threadIdx.x;
    if (i < N_NODES) {
        const float di = dinv[i];
        const float v = agg2[i] + di * di * z[i] + b2[0];
        out[i] = 1.0f / (1.0f + expf(-v));
    }
}

// ---------------- launcher ----------------

extern "C" void kernel_launch(void* const* d_in, const int* in_sizes, int n_in,
                              void* d_out, int out_size, void* d_ws, size_t ws_size,
                              hipStream_t stream) {
    const float* x   = (const float*)d_in[0];
    const int*   ei  = (const int*)d_in[1];       // [2, E]: src row then dst row
    const float* W1  = (const float*)d_in[2];
    const float* b1  = (const float*)d_in[3];
    const float* W2  = (const float*)d_in[4];
    const float* b2  = (const float*)d_in[5];
    float* out = (float*)d_out;

    const int* src = ei;
    const int* dst = ei + N_EDGES;

    char* ws = (char*)d_ws;
    // 16B-aligned offsets
    int*   degi = (int*)  (ws + 0);               // 400000 B
    float* dinv = (float*)(ws + 400128);          // 400000 B
    float* h    = (float*)(ws + 800256);          // 25.6 MB
    float* agg1 = (float*)(ws + 26400256);        // 25.6 MB (becomes h1 after relu)
    float* z    = (float*)(ws + 52000256);        // 400000 B
    float* agg2 = (float*)(ws + 52400256);        // 400000 B

    const int TB = 256;
    const int gN   = (N_NODES + TB - 1) / TB;             // 391
    const int gE   = (N_EDGES + TB - 1) / TB;             // 6250
    const int gNF  = (N_NODES * HID_DIM + TB - 1) / TB;   // 25000

    // degree + normalization
    k_init_deg <<<gN, TB, 0, stream>>>(degi);
    k_count_deg<<<gE, TB, 0, stream>>>(dst, degi);
    k_dinv     <<<gN, TB, 0, stream>>>(degi, dinv);

    // layer 1: GEMM (WMMA), scatter, relu+bias
    k_gemm1    <<<N_NODES / 32, 64, 0, stream>>>(x, W1, h);       // 3125 blocks
    k_zero     <<<gNF, TB, 0, stream>>>(agg1, N_NODES * HID_DIM);
    k_scatter1 <<<(N_EDGES * 16) / TB, TB, 0, stream>>>(src, dst, dinv, h, agg1); // 100000 blocks
    k_relu_bias<<<gNF, TB, 0, stream>>>(agg1, h, dinv, b1);

    // layer 2: matvec, scatter, sigmoid
    k_matvec2  <<<(N_NODES * 32 + TB - 1) / TB, TB, 0, stream>>>(agg1, W2, z);   // 12500 blocks
    k_zero     <<<gN, TB, 0, stream>>>(agg2, N_NODES);
    k_scatter2 <<<gE, TB, 0, stream>>>(src, dst, dinv, z, agg2);
    k_final    <<<gN, TB, 0, stream>>>(agg2, z, dinv, b2, out);
}